// _ConvBlock8_43018392436826
// MI455X (gfx1250) — compile-verified
//
#include <hip/hip_runtime.h>

typedef __attribute__((ext_vector_type(2))) float v2f;
typedef __attribute__((ext_vector_type(8))) float v8f;

// ---------------------------------------------------------------------------
// zero a float4-aligned region (accumulators must start at 0 every call)
// ---------------------------------------------------------------------------
__global__ __launch_bounds__(256) void zero4_kernel(float4* __restrict__ p, long long n4) {
    long long i = (long long)blockIdx.x * blockDim.x + threadIdx.x;
    if (i < n4) p[i] = make_float4(0.f, 0.f, 0.f, 0.f);
}

// ---------------------------------------------------------------------------
// out[N,64] = (RELU? max(A,0):A)[N,64] @ W[64,64]  using V_WMMA_F32_16X16X4_F32
// One wave32 computes a 16-row x 64-col tile: 16 K-steps x 4 N-tiles = 64 WMMAs.
// A-frag layout (32-bit 16x4): lanes 0-15 -> M=lane, K={0,1}; lanes 16-31 -> K={2,3}.
// B-frag layout (4x16, row striped across lanes): lane%16 = N, VGPR j -> K = 2*(lane/16)+j.
// C/D (16x16 f32): VGPR v, lane -> M = v + 8*(lane/16), N = lane%16.
// RELU is a compile-time template arg so the non-relu GEMM is a pure
// load+WMMA loop and the relu GEMM is a single v_max per element (no cndmask).
// ---------------------------------------------------------------------------
template <int RELU>
__global__ __launch_bounds__(256) void gemm64_wmma_kernel(
        const float* __restrict__ A, const float* __restrict__ W,
        float* __restrict__ out, int nrows)
{
    const int lane  = threadIdx.x & 31;
    const int wave  = threadIdx.x >> 5;
    const int tile  = blockIdx.x * 8 + wave;      // 16-row tile index
    const int r0    = tile * 16;
    if (r0 >= nrows) return;

    const int m     = lane & 15;                  // row within tile (A), col within N-tile (B/C)
    const int khalf = (lane >> 4) * 2;            // 0 or 2

    v8f c0 = {}, c1 = {}, c2 = {}, c3 = {};

    const int  arow_i = (r0 + m < nrows) ? (r0 + m) : (nrows - 1);
    const float* arow = A + (long long)arow_i * 64;

    #pragma unroll
    for (int k = 0; k < 64; k += 4) {
        v2f a;
        a.x = arow[k + khalf + 0];
        a.y = arow[k + khalf + 1];
        if (RELU) { a.x = fmaxf(a.x, 0.f); a.y = fmaxf(a.y, 0.f); }

        const float* wp = W + (long long)(k + khalf) * 64 + m;  // W[k+khalf][m + n*16]
        v2f b0, b1, b2, b3;
        b0.x = wp[0];   b0.y = wp[64];        // N-tile 0, K rows (k+khalf, k+khalf+1)
        b1.x = wp[16];  b1.y = wp[80];        // N-tile 1
        b2.x = wp[32];  b2.y = wp[96];        // N-tile 2
        b3.x = wp[48];  b3.y = wp[112];       // N-tile 3

        c0 = __builtin_amdgcn_wmma_f32_16x16x4_f32(false, a, false, b0, (short)0, c0, false, false);
        c1 = __builtin_amdgcn_wmma_f32_16x16x4_f32(false, a, false, b1, (short)0, c1, false, false);
        c2 = __builtin_amdgcn_wmma_f32_16x16x4_f32(false, a, false, b2, (short)0, c2, false, false);
        c3 = __builtin_amdgcn_wmma_f32_16x16x4_f32(false, a, false, b3, (short)0, c3, false, false);
    }

    #pragma unroll
    for (int v = 0; v < 8; ++v) {
        const int row = r0 + v + (lane >> 4) * 8;
        if (row < nrows) {
            float* orow = out + (long long)row * 64;
            orow[m +  0] = c0[v];
            orow[m + 16] = c1[v];
            orow[m + 32] = c2[v];
            orow[m + 48] = c3[v];
        }
    }
}

// ---------------------------------------------------------------------------
// Edge gather-scale-scatter: acc[dst[e]] += (RELU? max(h,0):h)[src[e]] * attr[e]
// 16 lanes per edge, float4 per lane, 4 non-returning global f32 atomics.
// ---------------------------------------------------------------------------
template <int RELU>
__global__ __launch_bounds__(256) void scatter_kernel(
        const float* __restrict__ h,
        const long long* __restrict__ src, const long long* __restrict__ dst,
        const float* __restrict__ attr, float* __restrict__ acc,
        long long E)
{
    long long tid = (long long)blockIdx.x * blockDim.x + threadIdx.x;
    long long e = tid >> 4;
    if (e >= E) return;
    int c = (int)(tid & 15) << 2;

    float a = attr[e];
    long long s = src[e];
    long long d = dst[e];

    float4 v = *(const float4*)(h + s * 64 + c);
    if (RELU) {
        v.x = fmaxf(v.x, 0.f); v.y = fmaxf(v.y, 0.f);
        v.z = fmaxf(v.z, 0.f); v.w = fmaxf(v.w, 0.f);
    }
    float* o = acc + d * 64 + c;
    atomicAdd(o + 0, v.x * a);
    atomicAdd(o + 1, v.y * a);
    atomicAdd(o + 2, v.z * a);
    atomicAdd(o + 3, v.w * a);
}

// ---------------------------------------------------------------------------
// host launch
// ---------------------------------------------------------------------------
extern "C" void kernel_launch(void* const* d_in, const int* in_sizes, int n_in,
                              void* d_out, int out_size, void* d_ws, size_t ws_size,
                              hipStream_t stream) {
    const float*     x     = (const float*)d_in[0];          // [N,64]
    const long long* eidx  = (const long long*)d_in[1];      // [2,E] int64
    const float*     eattr = (const float*)d_in[2];          // [E]
    const float*     w1    = (const float*)d_in[3];          // [64,64]
    const float*     w2    = (const float*)d_in[4];          // [64,64]
    const long long* usrc  = (const long long*)d_in[5];      // [Ep] int64
    const long long* udst  = (const long long*)d_in[6];      // [Ep] int64
    const float*     uattr = (const float*)d_in[7];          // [Ep]

    const int N  = in_sizes[0] / 64;
    const long long E  = in_sizes[1] / 2;
    const long long Ep = in_sizes[5];

    float* out  = (float*)d_out;
    float* bufA = (float*)d_ws;                   // GEMM output t       [N,64]
    float* bufB = bufA + (size_t)N * 64;          // scatter accumulator [N,64]

    const long long accN4 = (long long)N * 64 / 4;
    const long long outN4 = (long long)out_size / 4;
    const unsigned zbA = (unsigned)((accN4 + 255) / 256);
    const unsigned zbO = (unsigned)((outN4 + 255) / 256);

    const int tiles   = (N + 15) / 16;
    const unsigned gb = (unsigned)((tiles + 7) / 8);          // 8 waves/block
    const unsigned sbE  = (unsigned)((E  * 16 + 255) / 256);
    const unsigned sbEp = (unsigned)((Ep * 16 + 255) / 256);

    // layer 1: t1 = x @ w1 ; acc1 = scatter(t1)
    zero4_kernel<<<zbA, 256, 0, stream>>>((float4*)bufB, accN4);
    zero4_kernel<<<zbO, 256, 0, stream>>>((float4*)out, outN4);
    gemm64_wmma_kernel<0><<<gb, 256, 0, stream>>>(x, w1, bufA, N);
    scatter_kernel<0><<<sbE, 256, 0, stream>>>(bufA, eidx, eidx + E, eattr, bufB, E);

    // layer 2: t2 = relu(acc1) @ w2 ; acc2 = scatter(t2)   (acc2 reuses bufB)
    gemm64_wmma_kernel<1><<<gb, 256, 0, stream>>>(bufB, w2, bufA, N);
    zero4_kernel<<<zbA, 256, 0, stream>>>((float4*)bufB, accN4);
    scatter_kernel<0><<<sbE, 256, 0, stream>>>(bufA, eidx, eidx + E, eattr, bufB, E);

    // unpool: out += relu(acc2)[usrc] * uattr  scattered to fine nodes
    scatter_kernel<1><<<sbEp, 256, 0, stream>>>(bufB, usrc, udst, uattr, out, Ep);
}